// CurvatureLoss_81518479278835
// MI455X (gfx1250) — compile-verified
//
#include <hip/hip_runtime.h>
#include <math.h>
#include <stdint.h>

// Problem geometry (fixed by the reference)
#define BATCH 8
#define CH    4
#define IMH   1024
#define IMW   1024
#define TILE  64
#define HALO  4          // need 3; 4 keeps b128 async loads 16B-aligned
#define REG   72         // TILE + 2*HALO
#define RSTR  72         // LDS row stride in floats (72 mod 64 banks = 8 -> conflict-free)
#define RELEM (REG*RSTR) // 5184 floats per plane in LDS
#define NTHREADS 512

#define AS3 __attribute__((address_space(3)))

// LDS byte offset of a __shared__ object (generic -> local addrspacecast, 32-bit ptr)
__device__ __forceinline__ unsigned lds_off_of(const void* p) {
    return (unsigned)(unsigned long long)(AS3 const char*)p;
}

// CDNA5 async global->LDS, GVS mode: mem = SGPR64 base + signed VGPR byte offset.
// Moves 16B per lane (512B per wave per instruction), tracked by ASYNCcnt.
__device__ __forceinline__ void async_ld_b128(unsigned lds_off, unsigned gbyte_off,
                                              unsigned long long base) {
    asm volatile("global_load_async_to_lds_b128 %0, %1, %2"
                 :: "v"(lds_off), "v"(gbyte_off), "s"(base)
                 : "memory");
}

__device__ __forceinline__ void wait_async_zero() {
    asm volatile("s_wait_asynccnt 0" ::: "memory");
}

__device__ __forceinline__ int iclamp(int v, int lo, int hi) {
    return v < lo ? lo : (v > hi ? hi : v);
}

extern "C" __global__ void curv_init(float* ws) {
    int t = threadIdx.x;
    if (t < 48) ws[t] = 0.0f;
}

// Process one tile. INTERIOR=true: whole 72x72 region is inside the image
// (no clamps, no bounds predicates). INTERIOR=false: exact staged zero-pad
// semantics of the reference at image borders.
template <bool INTERIOR>
__device__ __forceinline__ void curv_tile(const float* __restrict__ pred,
                                          float* __restrict__ smem,
                                          int b, int x0, int y0, int tid,
                                          float* s_acc, float* c_acc) {
    float* stage = smem;               // 4 * RELEM  (async DMA target)
    float* prob  = smem + 4 * RELEM;   // 3 * RELEM
    float* lap   = smem;               // aliases stage plane 0 (safe after softmax)

    // ---- Stage 1: async-DMA the 4 channel sub-tiles (72x72 each) into LDS ----
    {
        const unsigned base_elem = (unsigned)b * (unsigned)(CH * IMH * IMW);
        const unsigned long long gbase = (unsigned long long)pred;
        // 18 float4 per 72-wide row, 72 rows, 4 channels = 5184 b128 transfers
        for (int i = tid; i < CH * REG * 18; i += NTHREADS) {
            int ch  = i / (REG * 18);
            int rem = i - ch * (REG * 18);
            int r   = rem / 18;
            int v4  = rem - r * 18;
            int gyy = y0 + r;
            int gxx = x0 + v4 * 4;
            if (!INTERIOR) {             // clamp to a legal, aligned address;
                gyy = iclamp(gyy, 0, IMH - 1);   // OOB positions zeroed logically later
                gxx = iclamp(gxx, 0, IMW - 4);
            }
            unsigned goff = (base_elem +
                             ((unsigned)ch * IMH + (unsigned)gyy) * (unsigned)IMW +
                             (unsigned)gxx) * 4u;
            unsigned loff = lds_off_of(&stage[ch * RELEM + r * RSTR + v4 * 4]);
            async_ld_b128(loff, goff, gbase);
        }
        wait_async_zero();      // each wave drains its own ASYNCcnt
        __syncthreads();        // then all waves' LDS writes are visible
    }

    // ---- Stage 2: softmax -> three prob fields (zero outside the image) ----
    if (INTERIOR) {
        // 4 pixels per iteration via 16B LDS accesses (planes are 16B-aligned)
        const float4* S0 = (const float4*)stage;
        const float4* S1 = (const float4*)(stage + RELEM);
        const float4* S2 = (const float4*)(stage + 2 * RELEM);
        const float4* S3 = (const float4*)(stage + 3 * RELEM);
        float4* P0 = (float4*)prob;
        float4* P1 = (float4*)(prob + RELEM);
        float4* P2 = (float4*)(prob + 2 * RELEM);
        for (int i4 = tid; i4 < RELEM / 4; i4 += NTHREADS) {
            float4 a0 = S0[i4], a1 = S1[i4], a2 = S2[i4], a3 = S3[i4];
            float4 o1, o12, o3;
#define SMAX1(k)                                                                  \
            { float m  = fmaxf(fmaxf(a0.k, a1.k), fmaxf(a2.k, a3.k));             \
              float e0 = __expf(a0.k - m), e1 = __expf(a1.k - m);                 \
              float e2 = __expf(a2.k - m), e3 = __expf(a3.k - m);                 \
              float inv = 1.0f / (e0 + e1 + e2 + e3);                             \
              o1.k = e1 * inv; o12.k = (e1 + e2) * inv; o3.k = e3 * inv; }
            SMAX1(x) SMAX1(y) SMAX1(z) SMAX1(w)
#undef SMAX1
            P0[i4] = o1; P1[i4] = o12; P2[i4] = o3;
        }
    } else {
        for (int i = tid; i < RELEM; i += NTHREADS) {
            int r = i / RSTR, c = i - (i / RSTR) * RSTR;
            int gyy = y0 + r, gxx = x0 + c;
            float p1 = 0.0f, p12 = 0.0f, p3 = 0.0f;
            if (((unsigned)gyy < (unsigned)IMH) && ((unsigned)gxx < (unsigned)IMW)) {
                float a0 = stage[i];
                float a1 = stage[RELEM + i];
                float a2 = stage[2 * RELEM + i];
                float a3 = stage[3 * RELEM + i];
                float m  = fmaxf(fmaxf(a0, a1), fmaxf(a2, a3));
                float e0 = __expf(a0 - m), e1 = __expf(a1 - m);
                float e2 = __expf(a2 - m), e3 = __expf(a3 - m);
                float inv = 1.0f / (e0 + e1 + e2 + e3);
                p1  = e1 * inv;
                p12 = (e1 + e2) * inv;
                p3  = e3 * inv;
            }
            prob[i]             = p1;
            prob[RELEM + i]     = p12;
            prob[2 * RELEM + i] = p3;
        }
    }

    // ---- Stage 3: per-field lap pass + fused gradient/curvature pass ----
    for (int f = 0; f < 3; ++f) {
        const float* pf = prob + f * RELEM;
        __syncthreads();   // prob ready / previous field's lap consumers done

        // laplacian on [2,69]^2 (68*68 = 4624) -- exactly the support the fused
        // curvature pass reads; zero at out-of-image positions (reference padding)
        for (int i = tid; i < 68 * 68; i += NTHREADS) {
            int r = 2 + i / 68, c = 2 + i % 68;
            int idx = r * RSTR + c;
            float l = 0.0f;
            bool inimg = true;
            if (!INTERIOR) {
                int gyy = y0 + r, gxx = x0 + c;
                inimg = ((unsigned)gyy < (unsigned)IMH) && ((unsigned)gxx < (unsigned)IMW);
            }
            if (inimg) {
                l = pf[idx - RSTR] + pf[idx + RSTR] + pf[idx - 1] + pf[idx + 1]
                    - 4.0f * pf[idx];
            }
            lap[idx] = l;
        }
        __syncthreads();

        // curvature on the 64x64 tile [4,67]; gx/gy expanded in-place from lap:
        //   gx(r,c) = lap(r,c+1)-lap(r,c-1),  gy(r,c) = lap(r+1,c)-lap(r-1,c)
        for (int i = tid; i < TILE * TILE; i += NTHREADS) {
            int r = HALO + (i >> 6), c = HALO + (i & 63);
            int idx = r * RSTR + c;
            float lC  = lap[idx];
            float lE1 = lap[idx + 1],        lE2 = lap[idx + 2];
            float lW1 = lap[idx - 1],        lW2 = lap[idx - 2];
            float lS1 = lap[idx + RSTR],     lS2 = lap[idx + 2 * RSTR];
            float lN1 = lap[idx - RSTR],     lN2 = lap[idx - 2 * RSTR];
            float gxc = lE1 - lW1;           // gx at (r,c)   -- (r,c) always in-image
            float gyc = lS1 - lN1;           // gy at (r,c)
            float gxe = lE2 - lC;            // gx at (r,c+1)
            float gxw = lC  - lW2;           // gx at (r,c-1)
            float gys = lS2 - lC;            // gy at (r+1,c)
            float gyn = lC  - lN2;           // gy at (r-1,c)
            if (!INTERIOR) {
                // a gx/gy sample is zero when its own position is outside the image
                int gyy = y0 + r, gxx = x0 + c;
                if (gxx + 1 >= IMW) gxe = 0.0f;
                if (gxx - 1 <  0)   gxw = 0.0f;
                if (gyy + 1 >= IMH) gys = 0.0f;
                if (gyy - 1 <  0)   gyn = 0.0f;
            }
            float hxx = 2.0f * gxc - gxe - gxw;
            float hyy = 2.0f * gyc - gys - gyn;
            float hxy = gxe - gxw;
            float opgy = 1.0f + gyc, opgx = 1.0f + gxc;
            float num  = hxx * opgy * opgy - 2.0f * hxy * gxc * gyc + hyy * opgx * opgx;
            float t    = 1.0f + gxc * gxc + gyc * gyc;
            float den  = 2.0f * t * __builtin_sqrtf(t);
            float curv = num / den;
            float neg  = fmaxf(-curv, 0.0f);
            s_acc[f] += neg;
            c_acc[f] += (neg != 0.0f) ? 1.0f : 0.0f;
        }
    }
}

extern "C" __global__ __launch_bounds__(NTHREADS)
void curv_main(const float* __restrict__ pred, float* __restrict__ ws) {
    __shared__ __align__(16) float smem[7 * RELEM];   // 145,152 B of the 320KB WGP LDS
    __shared__ float red[6];

    const int tid = threadIdx.x;
    const int b   = blockIdx.z;
    const int x0  = (int)blockIdx.x * TILE - HALO;
    const int y0  = (int)blockIdx.y * TILE - HALO;

    float s_acc[3] = {0.f, 0.f, 0.f};
    float c_acc[3] = {0.f, 0.f, 0.f};

    const bool interior = (x0 >= 0) && (y0 >= 0) &&
                          (x0 + REG <= IMW) && (y0 + REG <= IMH);
    if (interior) {
        curv_tile<true >(pred, smem, b, x0, y0, tid, s_acc, c_acc);
    } else {
        curv_tile<false>(pred, smem, b, x0, y0, tid, s_acc, c_acc);
    }

    // ---- Block reduction (wave32 shuffles) + global atomics ----
    __syncthreads();
    if (tid < 6) red[tid] = 0.0f;
    __syncthreads();
    const int lane = tid & 31;
    for (int k = 0; k < 3; ++k) {
        float v = s_acc[k];
        for (int o = 16; o > 0; o >>= 1) v += __shfl_down(v, o, 32);
        if (lane == 0) atomicAdd(&red[k], v);
        float w = c_acc[k];
        for (int o = 16; o > 0; o >>= 1) w += __shfl_down(w, o, 32);
        if (lane == 0) atomicAdd(&red[3 + k], w);
    }
    __syncthreads();
    // ws layout: s[f][b] at f*8+b, counts at 24 + f*8+b
    if (tid < 3)       atomicAdd(&ws[tid * 8 + b], red[tid]);
    else if (tid < 6)  atomicAdd(&ws[24 + (tid - 3) * 8 + b], red[tid]);
}

extern "C" __global__ void curv_fin(const float* __restrict__ ws, float* __restrict__ out) {
    if (threadIdx.x == 0) {
        float tot = 0.0f;
        for (int k = 0; k < 24; ++k) {
            float s = ws[k], c = ws[24 + k];
            if (c > 0.0f) tot += s / c;   // c >= 1 when > 0, so max(c,1) == c
        }
        out[0] = tot;   // LOSS_WEIGHT == 1.0
    }
}

extern "C" void kernel_launch(void* const* d_in, const int* in_sizes, int n_in,
                              void* d_out, int out_size, void* d_ws, size_t ws_size,
                              hipStream_t stream) {
    const float* pred = (const float*)d_in[0];   // (8,4,1024,1024) f32; target unused
    float* ws  = (float*)d_ws;                   // 48 floats of scratch
    float* out = (float*)d_out;                  // scalar f32

    hipLaunchKernelGGL(curv_init, dim3(1), dim3(64), 0, stream, ws);
    dim3 grid(IMW / TILE, IMH / TILE, BATCH);    // 16 x 16 x 8 blocks
    hipLaunchKernelGGL(curv_main, grid, dim3(NTHREADS), 0, stream, pred, ws);
    hipLaunchKernelGGL(curv_fin, dim3(1), dim3(32), 0, stream, ws, out);
}